// RoutePredictor_41996190221102
// MI455X (gfx1250) — compile-verified
//
#include <hip/hip_runtime.h>
#include <hip/hip_bf16.h>

typedef float v2f __attribute__((ext_vector_type(2)));
typedef float v8f __attribute__((ext_vector_type(8)));

// ---------------------------------------------------------------- utilities

__global__ void k_zero(float* __restrict__ p, long n) {
  long i = (long)blockIdx.x * blockDim.x + threadIdx.x;
  if (i < n) p[i] = 0.0f;
}

__global__ void k_degree(const int* __restrict__ col, float* __restrict__ deg, int E) {
  int e = blockIdx.x * blockDim.x + threadIdx.x;
  if (e < E) atomicAdd(deg + col[e], 1.0f);
}

// in-place: deg -> rsqrt(deg + 1)   (self-loop included)
__global__ void k_dinv(float* __restrict__ d, int N) {
  int i = blockIdx.x * blockDim.x + threadIdx.x;
  if (i < N) d[i] = rsqrtf(d[i] + 1.0f);
}

// ---------------------------------------------------------------- f32 WMMA GEMM
// C[N,M] = A[N,K] * B[K,M], row-major f32. One wave computes a 16x32 strip of C
// (two adjacent 16x16 tiles sharing the A fragment -> two independent WMMA
// dependency chains). Requires N%16==0, K%4==0, M%32==0 (true here).
//
// Layouts per CDNA5 ISA 7.12.2:
//   A 16x4 : lane%16 = M row; half-wave selects K={0,1} vs K={2,3}; 2 VGPRs.
//   B 4x16 : lane%16 = N col; half-wave selects K pair; 2 VGPRs.
//   C 16x16: VGPR r -> row r (lanes 0-15) / row r+8 (lanes 16-31), col = lane%16.
__global__ void k_gemm_wmma(const float* __restrict__ A, const float* __restrict__ B,
                            float* __restrict__ C, int N, int K, int M) {
  int tilesM2 = M >> 5;                                 // pairs of 16-wide tiles
  int wavesPerBlock = blockDim.x >> 5;
  int w = blockIdx.x * wavesPerBlock + (threadIdx.x >> 5);
  int totalW = (N >> 4) * tilesM2;
  if (w >= totalW) return;                              // wave-uniform: EXEC stays all-1s

  int lane = threadIdx.x & 31;
  int half = lane >> 4;                                 // K sub-phase (0 -> K0/K1, 1 -> K2/K3)
  int l16  = lane & 15;
  int tn = w / tilesM2;
  int tm = (w % tilesM2) << 1;                          // first of the two 16-wide tiles

  const float* aRow = A + (long)(tn * 16 + l16) * K + 2 * half;
  const float* b0   = B + (long)(2 * half) * M + tm * 16 + l16;
  const float* b1   = b0 + 16;

  v8f acc0 = {};
  v8f acc1 = {};
  for (int k = 0; k < K; k += 4) {
    v2f a = *(const v2f*)(aRow + k);                    // 8B-aligned (k+2*half even)
    v2f bb0; bb0.x = b0[(long)k * M]; bb0.y = b0[(long)(k + 1) * M];
    v2f bb1; bb1.x = b1[(long)k * M]; bb1.y = b1[(long)(k + 1) * M];
    acc0 = __builtin_amdgcn_wmma_f32_16x16x4_f32(false, a, false, bb0, (short)0, acc0, false, false);
    acc1 = __builtin_amdgcn_wmma_f32_16x16x4_f32(false, a, false, bb1, (short)0, acc1, false, false);
  }

  float* cBase = C + (long)(tn * 16 + 8 * half) * M + tm * 16 + l16;
#pragma unroll
  for (int r = 0; r < 8; ++r) {
    cBase[(long)r * M]      = acc0[r];
    cBase[(long)r * M + 16] = acc1[r];
  }
}

// ---------------------------------------------------------------- edge aggregation
// One wave per edge; each lane handles F/32 contiguous features.
// agg[col] += xw[row] * dinv[row]*dinv[col]   (float atomics, resolve in L2:
// xw and agg each fit in the 192MB global L2)
__global__ void k_spmm(const float* __restrict__ xw, const int* __restrict__ row,
                       const int* __restrict__ col, const float* __restrict__ dinv,
                       float* __restrict__ agg, int E, int F) {
  int wavesPerBlock = blockDim.x >> 5;
  int e = blockIdx.x * wavesPerBlock + (threadIdx.x >> 5);
  if (e >= E) return;
  int lane = threadIdx.x & 31;
  int r = row[e];
  int c = col[e];
  float norm = dinv[r] * dinv[c];
  const float* src = xw + (long)r * F;
  float* dst       = agg + (long)c * F;
  int per = F >> 5;                                     // 8 (F=256) or 4 (F=128)
  int f0 = lane * per;
  for (int j = 0; j < per; j += 4) {
    float4 v = *(const float4*)(src + f0 + j);
    atomicAdd(dst + f0 + j + 0, v.x * norm);
    atomicAdd(dst + f0 + j + 1, v.y * norm);
    atomicAdd(dst + f0 + j + 2, v.z * norm);
    atomicAdd(dst + f0 + j + 3, v.w * norm);
  }
}

// ---------------------------------------------------------------- finalize
__device__ __forceinline__ float gelu_tanh(float x) {
  const float c = 0.7978845608028654f;                  // sqrt(2/pi)
  float x3 = x * x * x;
  return 0.5f * x * (1.0f + tanhf(c * (x + 0.044715f * x3)));
}

// h = gelu(agg + xw * dinv^2 + b)   (in place into agg)
__global__ void k_finalize_gelu(float* __restrict__ agg, const float* __restrict__ xw,
                                const float* __restrict__ dinv, const float* __restrict__ bias,
                                long total, int F) {
  long i = (long)blockIdx.x * blockDim.x + threadIdx.x;
  if (i >= total) return;
  long n = i / F;
  int  f = (int)(i - n * F);
  float di = dinv[n];
  float v = agg[i] + xw[i] * di * di + bias[f];
  agg[i] = gelu_tanh(v);
}

// out += hw * dinv^2 + b    (out already holds the atomic aggregation)
__global__ void k_finalize_out(float* __restrict__ out, const float* __restrict__ hw,
                               const float* __restrict__ dinv, const float* __restrict__ bias,
                               long total, int F) {
  long i = (long)blockIdx.x * blockDim.x + threadIdx.x;
  if (i >= total) return;
  long n = i / F;
  int  f = (int)(i - n * F);
  float di = dinv[n];
  out[i] = out[i] + hw[i] * di * di + bias[f];
}

// ---------------------------------------------------------------- launch

extern "C" void kernel_launch(void* const* d_in, const int* in_sizes, int n_in,
                              void* d_out, int out_size, void* d_ws, size_t ws_size,
                              hipStream_t stream) {
  const float* x  = (const float*)d_in[0];
  const int*   ei = (const int*)d_in[1];
  const float* W1 = (const float*)d_in[2];
  const float* b1 = (const float*)d_in[3];
  const float* W2 = (const float*)d_in[4];
  const float* b2 = (const float*)d_in[5];
  float* out = (float*)d_out;

  int H    = in_sizes[3];                 // 256
  int Fin  = in_sizes[2] / H;             // 128
  int N    = in_sizes[0] / Fin;           // 100000
  int Fout = in_sizes[5];                 // 128
  int E    = in_sizes[1] / 2;             // 1.6M
  const int* row = ei;
  const int* col = ei + E;

  // workspace layout
  char* ws = (char*)d_ws;
  float* dinv = (float*)ws;                                            // N floats
  size_t off  = ((size_t)N * sizeof(float) + 511) & ~(size_t)511;
  float* xw1  = (float*)(ws + off);                                    // N*H floats
  float* agg1 = xw1 + (size_t)N * H;                                   // N*H floats (-> h)
  float* hw2  = xw1;                                                   // reuse after layer1

  long NH = (long)N * H;
  long NO = (long)N * Fout;

  const int T = 256;
  // zero deg, agg1, out
  k_zero<<<(int)((N  + T - 1) / T), T, 0, stream>>>(dinv, N);
  k_zero<<<(int)((NH + T - 1) / T), T, 0, stream>>>(agg1, NH);
  k_zero<<<(int)((NO + T - 1) / T), T, 0, stream>>>(out, NO);

  // degree + dinv
  k_degree<<<(E + T - 1) / T, T, 0, stream>>>(col, dinv, E);
  k_dinv<<<(N + T - 1) / T, T, 0, stream>>>(dinv, N);

  int wavesPerBlock = T / 32;

  // layer 1: xw1 = x @ W1
  {
    int totalW = (N / 16) * (H / 32);
    int blocks = (totalW + wavesPerBlock - 1) / wavesPerBlock;
    k_gemm_wmma<<<blocks, T, 0, stream>>>(x, W1, xw1, N, Fin, H);
  }
  // aggregate into agg1
  {
    int blocks = (E + wavesPerBlock - 1) / wavesPerBlock;
    k_spmm<<<blocks, T, 0, stream>>>(xw1, row, col, dinv, agg1, E, H);
  }
  // h = gelu(agg1 + xw1/deg + b1)  (in place in agg1)
  k_finalize_gelu<<<(int)((NH + T - 1) / T), T, 0, stream>>>(agg1, xw1, dinv, b1, NH, H);

  // layer 2: hw2 = h @ W2
  {
    int totalW = (N / 16) * (Fout / 32);
    int blocks = (totalW + wavesPerBlock - 1) / wavesPerBlock;
    k_gemm_wmma<<<blocks, T, 0, stream>>>(agg1, W2, hw2, N, H, Fout);
  }
  // aggregate directly into out
  {
    int blocks = (E + wavesPerBlock - 1) / wavesPerBlock;
    k_spmm<<<blocks, T, 0, stream>>>(hw2, row, col, dinv, out, E, Fout);
  }
  // out += hw2/deg + b2
  k_finalize_out<<<(int)((NO + T - 1) / T), T, 0, stream>>>(out, hw2, dinv, b2, NO, Fout);
}